// TorchNeighborList_26156350832801
// MI455X (gfx1250) — compile-verified
//
#include <hip/hip_runtime.h>

typedef __attribute__((ext_vector_type(2))) float v2f;
typedef __attribute__((ext_vector_type(8))) float v8f;
typedef __attribute__((ext_vector_type(4))) int   v4i;

#define N_ATOMS     8192
#define MAX_PAIRS   1048576
#define CUTOFF2     25.0f            // CUTOFF^2
#define NTILES      (N_ATOMS / 16)   // 512 column tiles total
#define CHUNK_ATOMS 2048             // atoms staged in LDS per step (24 KB)
#define NCHUNKS     (N_ATOMS / CHUNK_ATOMS)
#define CHUNK_TILES (CHUNK_ATOMS / 16)

// --- gfx1250 async global->LDS copy (ASYNCcnt path), with safe fallbacks ----
#if defined(__has_builtin)
#  if __has_builtin(__builtin_amdgcn_global_load_async_to_lds_b128)
#    define HAVE_ASYNC_LDS 1
#  endif
#  if __has_builtin(__builtin_amdgcn_s_wait_asynccnt)
#    define HAVE_WAIT_ASYNC 1
#  endif
#endif
#ifndef HAVE_ASYNC_LDS
#  define HAVE_ASYNC_LDS 0
#endif
#ifndef HAVE_WAIT_ASYNC
#  define HAVE_WAIT_ASYNC 0
#endif

// Builtin signature (from hipcc diagnostic): arg0 = int4 AS(1)* (global src),
// arg1 = int4 AS(3)* (LDS dst), arg2/arg3 = imm offset / imm cpol.
typedef __attribute__((address_space(1))) v4i as1_v4i;
typedef __attribute__((address_space(3))) v4i as3_v4i;

__device__ __forceinline__ unsigned ballot32(bool p) {
    return (unsigned)__ballot(p ? 1 : 0);
}

// Stage CHUNK_ATOMS atoms (24576 bytes) into LDS, block-cooperatively.
// 1536 x b128 transfers: 256 threads x 6 each.
__device__ __forceinline__ void stage_chunk(const float* __restrict__ pos,
                                            float* lds, int chunkBase) {
    const int t = threadIdx.x; // 0..255
    const char* g = (const char*)(pos + (size_t)chunkBase * 3);
    char* l = (char*)lds;
#if HAVE_ASYNC_LDS
#pragma unroll
    for (int k = 0; k < 6; ++k) {
        __builtin_amdgcn_global_load_async_to_lds_b128(
            (as1_v4i*)(void*)(g + (t * 16 + k * 4096)),
            (as3_v4i*)(void*)(l + (t * 16 + k * 4096)),
            0, 0);
    }
#  if HAVE_WAIT_ASYNC
    __builtin_amdgcn_s_wait_asynccnt(0);
#  else
    asm volatile("s_wait_asynccnt 0x0" ::: "memory");
#  endif
#else
    const float4* g4 = (const float4*)g;
    float4* l4 = (float4*)l;
#pragma unroll
    for (int k = 0; k < 6; ++k) l4[t + k * 256] = g4[t + k * 256];
#endif
}

// ---------------------------------------------------------------------------
// Kernel 0: zero the whole output (pairs + scales + ds). Padding semantics:
// pairs=(0,0), scales=0, ds=0.0. Also makes graph replays deterministic.
// ---------------------------------------------------------------------------
__global__ void nl_zero(int* __restrict__ out, int n) {
    int t = blockIdx.x * blockDim.x + threadIdx.x;
    if (t < n) out[t] = 0;
}

// ---------------------------------------------------------------------------
// d2 tile via one V_WMMA_F32_16X16X4_F32 (reference gram trick):
//   A row m = (-2x_i, -2y_i, -2z_i, 1)
//   B col n = ( x_j,   y_j,   z_j,  |x_j|^2 )
//   C[m][n] = |x_i|^2  ->  D = d2
// C/D layout (ISA 7.12.2): VGPR v, lanes 0-15 -> (M=v, N=lane),
//                                   lanes 16-31 -> (M=v+8, N=lane-16)
// ---------------------------------------------------------------------------

// Kernel 1: per-row neighbor counts (no atomics, fully deterministic).
__global__ void nl_count(const float* __restrict__ pos, int* __restrict__ counts) {
    __shared__ float s_pos[CHUNK_ATOMS * 3];

    const int lane    = threadIdx.x & 31;
    const int wave    = threadIdx.x >> 5;
    const int rowTile = blockIdx.x * 8 + wave; // 0..511
    const int i0      = rowTile * 16;
    const int m       = lane & 15;

    const int ia = i0 + m;
    const float xi = pos[ia * 3 + 0];
    const float yi = pos[ia * 3 + 1];
    const float zi = pos[ia * 3 + 2];
    const float sqi = xi * xi + yi * yi + zi * zi;

    v2f a;
    if (lane < 16) { a.x = -2.0f * xi; a.y = -2.0f * yi; }
    else           { a.x = -2.0f * zi; a.y = 1.0f;       }

    v8f c;
#pragma unroll
    for (int v = 0; v < 8; ++v)
        c[v] = __shfl(sqi, (lane < 16) ? v : (v + 8), 32);

    int count = 0; // lane m (<16) accumulates count for row i0+m

    for (int ch = 0; ch < NCHUNKS; ++ch) {
        __syncthreads(); // previous chunk fully consumed
        stage_chunk(pos, s_pos, ch * CHUNK_ATOMS);
        __syncthreads(); // chunk visible to all waves

        for (int jt = 0; jt < CHUNK_TILES; ++jt) {
            const int j0 = ch * CHUNK_ATOMS + jt * 16;
            const int jl = jt * 16 + m; // atom index within LDS chunk
            const float xj = s_pos[jl * 3 + 0];
            const float yj = s_pos[jl * 3 + 1];
            const float zj = s_pos[jl * 3 + 2];
            const float sqj = xj * xj + yj * yj + zj * zj;

            v2f b;
            if (lane < 16) { b.x = xj; b.y = yj;  }
            else           { b.x = zj; b.y = sqj; }

            v8f d = __builtin_amdgcn_wmma_f32_16x16x4_f32(
                false, a, false, b, (short)0, c, false, false);

#pragma unroll
            for (int v = 0; v < 8; ++v) {
                const int row = (lane < 16) ? v : (v + 8);
                const int i = i0 + row;
                const int j = j0 + m;
                const bool pass = (d[v] < CUTOFF2) && (i != j);
                const unsigned mk = ballot32(pass); // [15:0]=row v, [31:16]=row v+8
                const int clo = __popc(mk & 0xFFFFu);
                const int chi = __popc(mk >> 16);
                if (lane == v)     count += clo;
                if (lane == v + 8) count += chi;
            }
        }
    }
    if (lane < 16) counts[i0 + lane] = count;
}

// Kernel 2: exclusive prefix sum over 8192 counts -> row offsets (+ total).
__global__ void nl_scan(const int* __restrict__ counts, int* __restrict__ offsets) {
    __shared__ int sums[256];
    const int t = threadIdx.x;
    int local[32];
    int s = 0;
#pragma unroll
    for (int k = 0; k < 32; ++k) { local[k] = s; s += counts[t * 32 + k]; }
    sums[t] = s;
    __syncthreads();
    for (int off = 1; off < 256; off <<= 1) {
        int v = (t >= off) ? sums[t - off] : 0;
        __syncthreads();
        sums[t] += v;
        __syncthreads();
    }
    const int base = (t == 0) ? 0 : sums[t - 1];
#pragma unroll
    for (int k = 0; k < 32; ++k) offsets[t * 32 + k] = base + local[k];
    if (t == 255) offsets[N_ATOMS] = sums[255];
}

// Kernel 3: ordered emission. Row cursors live in lanes 0..15 and advance as
// column tiles are swept left-to-right, so the global write order exactly
// matches jnp.nonzero's row-major order.
__global__ void nl_emit(const float* __restrict__ pos,
                        const int* __restrict__ offsets,
                        int* __restrict__ pairs,
                        int* __restrict__ scales,
                        float* __restrict__ ds) {
    __shared__ float s_pos[CHUNK_ATOMS * 3];

    const int lane    = threadIdx.x & 31;
    const int wave    = threadIdx.x >> 5;
    const int rowTile = blockIdx.x * 8 + wave;
    const int i0      = rowTile * 16;
    const int m       = lane & 15;

    const int ia = i0 + m;
    const float xi = pos[ia * 3 + 0];
    const float yi = pos[ia * 3 + 1];
    const float zi = pos[ia * 3 + 2];
    const float sqi = xi * xi + yi * yi + zi * zi;

    v2f a;
    if (lane < 16) { a.x = -2.0f * xi; a.y = -2.0f * yi; }
    else           { a.x = -2.0f * zi; a.y = 1.0f;       }

    v8f c;
#pragma unroll
    for (int v = 0; v < 8; ++v)
        c[v] = __shfl(sqi, (lane < 16) ? v : (v + 8), 32);

    int cur = (lane < 16) ? offsets[i0 + lane] : 0; // write cursor for row i0+lane

    const unsigned lowmask = (1u << lane) - 1u;
    const unsigned halfsel = (lane < 16) ? 0xFFFFu : 0xFFFF0000u;

    for (int ch = 0; ch < NCHUNKS; ++ch) {
        __syncthreads();
        stage_chunk(pos, s_pos, ch * CHUNK_ATOMS);
        __syncthreads();

        for (int jt = 0; jt < CHUNK_TILES; ++jt) {
            const int j0 = ch * CHUNK_ATOMS + jt * 16;
            const int jl = jt * 16 + m;
            const float xj = s_pos[jl * 3 + 0];
            const float yj = s_pos[jl * 3 + 1];
            const float zj = s_pos[jl * 3 + 2];
            const float sqj = xj * xj + yj * yj + zj * zj;

            v2f b;
            if (lane < 16) { b.x = xj; b.y = yj;  }
            else           { b.x = zj; b.y = sqj; }

            v8f d = __builtin_amdgcn_wmma_f32_16x16x4_f32(
                false, a, false, b, (short)0, c, false, false);

#pragma unroll
            for (int v = 0; v < 8; ++v) {
                const int row = (lane < 16) ? v : (v + 8);
                const int i = i0 + row;
                const int j = j0 + m;
                const bool pass = (d[v] < CUTOFF2) && (i != j);
                const unsigned mk = ballot32(pass);
                const int rank = __popc(mk & lowmask & halfsel);

                // uniform control flow: shuffles before divergence
                const int   base = __shfl(cur, row, 32);
                const float xr   = __shfl(xi, row, 32);
                const float yr   = __shfl(yi, row, 32);
                const float zr   = __shfl(zi, row, 32);

                if (pass) {
                    const int idx = base + rank;
                    if (idx < MAX_PAIRS) {
                        pairs[2 * idx + 0] = i;
                        pairs[2 * idx + 1] = j;
                        scales[idx] = (i < j) ? 1 : 0;
                        const float dx = xj - xr;
                        const float dy = yj - yr;
                        const float dz = zj - zr;
                        ds[idx] = sqrtf(dx * dx + dy * dy + dz * dz);
                    }
                }

                const int clo = __popc(mk & 0xFFFFu);
                const int chi = __popc(mk >> 16);
                if (lane == v)     cur += clo;
                if (lane == v + 8) cur += chi;
            }
        }
    }
}

// ---------------------------------------------------------------------------
extern "C" void kernel_launch(void* const* d_in, const int* in_sizes, int n_in,
                              void* d_out, int out_size, void* d_ws, size_t ws_size,
                              hipStream_t stream) {
    (void)in_sizes; (void)n_in; (void)out_size; (void)ws_size;

    const float* pos = (const float*)d_in[0];

    int* counts  = (int*)d_ws;            // 8192 ints
    int* offsets = counts + N_ATOMS;      // 8193 ints

    int*   pairs  = (int*)d_out;                  // [MAX_PAIRS, 2] int32
    int*   scales = pairs + 2 * MAX_PAIRS;        // [MAX_PAIRS]    int32
    float* ds     = (float*)(scales + MAX_PAIRS); // [MAX_PAIRS]    float32

    const int total_out_ints = 4 * MAX_PAIRS;
    nl_zero<<<total_out_ints / 256, 256, 0, stream>>>((int*)d_out, total_out_ints);
    nl_count<<<NTILES / 8, 256, 0, stream>>>(pos, counts);
    nl_scan<<<1, 256, 0, stream>>>(counts, offsets);
    nl_emit<<<NTILES / 8, 256, 0, stream>>>(pos, offsets, pairs, scales, ds);
}